// TransOpt_43946105373030
// MI455X (gfx1250) — compile-verified
//
#include <hip/hip_runtime.h>
#include <hip/hip_bf16.h>

// CDNA5 f32 WMMA fragment types (wave32): A/B 16x4 & 4x16 f32 -> 2 VGPRs, C/D 16x16 f32 -> 8 VGPRs
typedef __attribute__((ext_vector_type(2))) float v2f;
typedef __attribute__((ext_vector_type(8))) float v8f;

#define LDN 68              // padded LDS row stride (floats): +4 banks/row -> conflict-free strided access
#define NB  64              // matrix dimension
#define MSZ (NB * LDN)      // floats per matrix buffer (4352)
#define TPB 512             // 16 waves: 8 per batch element
#define BPB 2               // batch elements per block (share one Psi sweep)

// Two adjacent 16x16 tiles (tm, tn0) and (tm, tn0+1) of C = A(64x64) * B(64x64).
// A is read row-major; B is read from its TRANSPOSED copy Bt, so every fragment
// (A-pair and B-pair) is a contiguous 8-byte LDS load into an adjacent VGPR pair —
// no register re-marshalling movs before the WMMAs.
__device__ __forceinline__ void mm_tile2(const float* __restrict__ A,
                                         const float* __restrict__ Bt,
                                         int tm, int tn0, int lane,
                                         v8f& o0, v8f& o1) {
    const int r  = lane & 15;            // row (A) / col (B) within tile
    const int kh = (lane >> 4) << 1;     // K sub-pair: lanes 0-15 -> K{0,1}, lanes 16-31 -> K{2,3}

    v2f af[16], b0[16], b1[16];
    const float* arow = A  + (tm * 16 + r)       * LDN + kh;  // A[row][k..k+1]
    const float* brow = Bt + (tn0 * 16 + r)      * LDN + kh;  // Bt[col][k..k+1] = B[k..k+1][col]
    const float* crow = Bt + (tn0 * 16 + 16 + r) * LDN + kh;

    #pragma unroll
    for (int s = 0; s < 16; ++s) {
        af[s] = *(const v2f*)(arow + s * 4);
        b0[s] = *(const v2f*)(brow + s * 4);
        b1[s] = *(const v2f*)(crow + s * 4);
    }
    asm volatile("" ::: "memory");       // keep the whole load batch ahead of the WMMA chain

    v8f c00 = {0.f, 0.f, 0.f, 0.f, 0.f, 0.f, 0.f, 0.f};
    v8f c01 = c00, c10 = c00, c11 = c00; // 4 independent accumulator chains
    #pragma unroll
    for (int s = 0; s < 16; s += 2) {
        c00 = __builtin_amdgcn_wmma_f32_16x16x4_f32(false, af[s],     false, b0[s],
                                                    (short)0, c00, false, false);
        c10 = __builtin_amdgcn_wmma_f32_16x16x4_f32(false, af[s],     false, b1[s],
                                                    (short)0, c10, false, false);
        c01 = __builtin_amdgcn_wmma_f32_16x16x4_f32(false, af[s + 1], false, b0[s + 1],
                                                    (short)0, c01, false, false);
        c11 = __builtin_amdgcn_wmma_f32_16x16x4_f32(false, af[s + 1], false, b1[s + 1],
                                                    (short)0, c11, false, false);
    }
    o0 = c00 + c01;
    o1 = c10 + c11;
}

// Scatter a C/D tile to LDS in BOTH layouts (normal + transposed), with scale and +I.
// Transposed side: 8 consecutive floats per lane -> wide ds_store merges.
__device__ __forceinline__ void store_tile(float* __restrict__ dstN,
                                           float* __restrict__ dstT,
                                           int tm, int tn, int lane,
                                           const v8f acc, float scale, bool addI) {
    const int c     = lane & 15;
    const int rbase = (lane >> 4) * 8;   // C/D layout: VGPR v -> row v (+8 for high lanes)
    const int gc    = tn * 16 + c;
    float vals[8];
    #pragma unroll
    for (int v = 0; v < 8; ++v) {
        const int gr = tm * 16 + rbase + v;
        float val = acc[v] * scale;
        if (addI && (gr == gc)) val += 1.0f;
        vals[v] = val;
        dstN[gr * LDN + gc] = val;       // normal: strided (2addr_b32 pairs)
    }
    float* tp = dstT + gc * LDN + tm * 16 + rbase;
    #pragma unroll
    for (int v = 0; v < 8; ++v)
        tp[v] = vals[v];                 // transposed: contiguous -> b64/b128 stores
}

__global__ __launch_bounds__(TPB)
void expm_apply_kernel(const float* __restrict__ input_z,   // [B,64]
                       const float* __restrict__ coeff,     // [B,16]
                       const float* __restrict__ Psi,       // [4096,16]
                       const float* __restrict__ noise,     // [B,64]
                       float* __restrict__ out) {           // [B,64]
    extern __shared__ float smem[];
    float* Xs  = smem;                   // [BPB][MSZ]  X = A / 2^6 (row-major, A-operand only)
    float* Pn0 = smem + 1 * BPB * MSZ;   // ping, normal
    float* Pn1 = smem + 2 * BPB * MSZ;   // pong, normal
    float* Pt0 = smem + 3 * BPB * MSZ;   // ping, transposed
    float* Pt1 = smem + 4 * BPB * MSZ;   // pong, transposed
    float* xv  = smem + 5 * BPB * MSZ;   // [BPB][64]
    float* cs  = xv + BPB * NB;          // [BPB][16]

    const int b0   = blockIdx.x * BPB;
    const int tid  = threadIdx.x;
    const int lane = tid & 31;
    const int wave = tid >> 5;
    const int bw   = wave >> 3;          // batch element this wave works on (0/1)
    const int w8   = wave & 7;
    const int tm   = w8 >> 1;            // 0..3
    const int tn0  = (w8 & 1) * 2;       // 0 or 2 (owns tiles tn0, tn0+1)

    if (tid < BPB * 16) cs[tid] = coeff[b0 * 16 + tid];
    if (tid < BPB * NB) xv[tid] = input_z[b0 * NB + tid];
    __syncthreads();

    // Coefficients for both batch elements into registers.
    float c0r[16], c1r[16];
    #pragma unroll
    for (int m = 0; m < 16; ++m) { c0r[m] = cs[m]; c1r[m] = cs[16 + m]; }

    // ---- One Psi sweep feeds both batches: X = A/64, P0 = I + X/8 (k=8 Horner step) ----
    for (int f = tid; f < NB * NB; f += TPB) {
        const float4* p = (const float4*)(Psi + f * 16);
        const float4 p0 = p[0], p1 = p[1], p2 = p[2], p3 = p[3];
        float s0 = p0.x*c0r[0]  + p0.y*c0r[1]  + p0.z*c0r[2]  + p0.w*c0r[3]
                 + p1.x*c0r[4]  + p1.y*c0r[5]  + p1.z*c0r[6]  + p1.w*c0r[7]
                 + p2.x*c0r[8]  + p2.y*c0r[9]  + p2.z*c0r[10] + p2.w*c0r[11]
                 + p3.x*c0r[12] + p3.y*c0r[13] + p3.z*c0r[14] + p3.w*c0r[15];
        float s1 = p0.x*c1r[0]  + p0.y*c1r[1]  + p0.z*c1r[2]  + p0.w*c1r[3]
                 + p1.x*c1r[4]  + p1.y*c1r[5]  + p1.z*c1r[6]  + p1.w*c1r[7]
                 + p2.x*c1r[8]  + p2.y*c1r[9]  + p2.z*c1r[10] + p2.w*c1r[11]
                 + p3.x*c1r[12] + p3.y*c1r[13] + p3.z*c1r[14] + p3.w*c1r[15];
        const int i = f & 63, j = f >> 6;
        const float x0 = s0 * (1.0f / 64.0f);
        const float x1 = s1 * (1.0f / 64.0f);
        const float d  = (i == j) ? 1.0f : 0.0f;
        const float q0 = x0 * 0.125f + d;
        const float q1 = x1 * 0.125f + d;
        Xs [i * LDN + j]       = x0;
        Xs [MSZ + i * LDN + j] = x1;
        Pn0[i * LDN + j]       = q0;     // P0
        Pn0[MSZ + i * LDN + j] = q1;
        Pt0[j * LDN + i]       = q0;     // P0^T
        Pt0[MSZ + j * LDN + i] = q1;
    }
    __syncthreads();

    const float* Xw = Xs + bw * MSZ;
    float* curN = Pn0 + bw * MSZ;  float* curT = Pt0 + bw * MSZ;
    float* nxtN = Pn1 + bw * MSZ;  float* nxtT = Pt1 + bw * MSZ;

    // ---- Horner: P <- I + (X @ P)/k, k = 7..1 (fully unrolled, constant reciprocals) ----
    const float invk[7] = {1.f/7.f, 1.f/6.f, 1.f/5.f, 0.25f, 1.f/3.f, 0.5f, 1.f};
    #pragma unroll
    for (int it = 0; it < 7; ++it) {
        v8f o0, o1;
        mm_tile2(Xw, curT, tm, tn0, lane, o0, o1);
        store_tile(nxtN, nxtT, tm, tn0,     lane, o0, invk[it], true);
        store_tile(nxtN, nxtT, tm, tn0 + 1, lane, o1, invk[it], true);
        float* t;
        t = curN; curN = nxtN; nxtN = t;
        t = curT; curT = nxtT; nxtT = t;
        __syncthreads();
    }

    // ---- Squaring: T <- T @ T, 6 times (undo 1/2^6 scaling) ----
    #pragma unroll 1
    for (int t = 0; t < 6; ++t) {
        v8f o0, o1;
        mm_tile2(curN, curT, tm, tn0, lane, o0, o1);
        store_tile(nxtN, nxtT, tm, tn0,     lane, o0, 1.0f, false);
        store_tile(nxtN, nxtT, tm, tn0 + 1, lane, o1, 1.0f, false);
        float* tp;
        tp = curN; curN = nxtN; nxtN = tp;
        tp = curT; curT = nxtT; nxtT = tp;
        __syncthreads();
    }

    // After 13 swaps (odd) the result lives in the 'pong' buffers (Pn1).
    // ---- z = T @ x + noise ----
    if (tid < BPB * NB) {
        const int bb = tid >> 6, i = tid & 63;
        const float* T = Pn1 + bb * MSZ;
        const float* x = xv + bb * NB;
        float s = 0.0f;
        #pragma unroll
        for (int j = 0; j < NB; ++j)
            s = fmaf(T[i * LDN + j], x[j], s);
        out[(b0 + bb) * NB + i] = s + noise[(b0 + bb) * NB + i];
    }
}

extern "C" void kernel_launch(void* const* d_in, const int* in_sizes, int n_in,
                              void* d_out, int out_size, void* d_ws, size_t ws_size,
                              hipStream_t stream) {
    const float* input_z = (const float*)d_in[0];  // [8192,64]
    const float* coeff   = (const float*)d_in[1];  // [8192,16]
    const float* Psi     = (const float*)d_in[2];  // [4096,16]
    const float* noise   = (const float*)d_in[3];  // [8192,64]
    float* out = (float*)d_out;

    const int B = in_sizes[0] / 64;                // 8192
    const size_t shmem = (size_t)(5 * BPB * MSZ + BPB * (NB + 16)) * sizeof(float);
    expm_apply_kernel<<<B / BPB, TPB, shmem, stream>>>(input_z, coeff, Psi, noise, out);
}